// Q_network_59399397704417
// MI455X (gfx1250) — compile-verified
//
#include <hip/hip_runtime.h>
#include <math.h>

typedef __attribute__((ext_vector_type(2))) float v2f;
typedef __attribute__((ext_vector_type(8))) float v8f;

#define NTHREADS 256
#define NW 8   // waves per block (wave32)

// ---------------- LDS layout (float offsets) ----------------
#define LD_SACT 145               // 16 x [128 states | 16 actions | pad]
#define OFF_SACT 0                // 2320
#define OFF_SE   2320             // 16 x 65
#define OFF_SAE  3360
#define OFF_KK   4400
#define OFF_QQ   5440
#define OFF_VV   6480
#define OFF_X    7520
#define OFF_FF   8560
#define OFF_X2   9600
#define OFF_C2   10640
#define OFF_QC   11680
#define OFF_QV   12720            // 16 x 17
#define OFF_SC   12992            // 4*12*11 = 528 (score, already /4)
#define OFF_WT   13520            // 528 (attention weights)
#define OFF_HW   14048            // 132 hard gate
#define OFF_LG   14180            // 144 x 2 logits (padded rows)
#define OFF_BIG  14468            // union scratch
#define BIG_SZ   32784            // max(qk 144x129 + hg 144x65, h1 16x2049, cin+cout 2*16*129)
#define SMEM_FLOATS (OFF_BIG + BIG_SZ)

struct Params {
  const float *states, *actions;
  const float *fn_g, *fn_b;
  const float *se_W, *se_b, *se_g, *se_bb;
  const float *sae_W, *sae_b, *sae_g, *sae_bb;
  const float *k_W, *k_b, *q_W, *q_b, *v_W, *v_b;
  const float *avln_g, *avln_b;
  const float *ff1_W, *ff1_b, *ff_g, *ff_bb, *ff2_W, *ff2_b;
  const float *avlln_g, *avlln_b;
  const float *ha1_W, *ha1_b, *ha2_W, *ha2_b;
  const float *c1_W, *c1_b, *c1_g, *c1_bb;
  const float *c2_W, *c2_b, *c2_g, *c2_bb;
  const float *qln_g, *qln_b, *qv_W, *qv_b;
  float *outQ, *outWf, *outScore;
};

static __device__ inline float wave_sum(float v) {
  #pragma unroll
  for (int m = 16; m >= 1; m >>= 1) v += __shfl_xor(v, m, 32);
  return v;
}

static __device__ inline float gelu_exact(float x) {
  return 0.5f * x * (1.0f + erff(x * 0.70710678118654752f));
}

// ---------------- threefry2x32 (JAX-compatible, key(42) = [0,42]) ----------------
static __device__ inline unsigned rotl32(unsigned x, unsigned r) {
  return (x << r) | (x >> (32u - r));
}

static __device__ inline void threefry2x32(unsigned k0, unsigned k1,
                                           unsigned x0, unsigned x1,
                                           unsigned &o0, unsigned &o1) {
  unsigned ks0 = k0, ks1 = k1, ks2 = k0 ^ k1 ^ 0x1BD11BDAu;
  const unsigned ra[4] = {13u, 15u, 26u, 6u};
  const unsigned rb[4] = {17u, 29u, 16u, 24u};
  x0 += ks0; x1 += ks1;
  #pragma unroll
  for (int i = 0; i < 4; ++i) { x0 += x1; x1 = rotl32(x1, ra[i]); x1 ^= x0; }
  x0 += ks1; x1 += ks2 + 1u;
  #pragma unroll
  for (int i = 0; i < 4; ++i) { x0 += x1; x1 = rotl32(x1, rb[i]); x1 ^= x0; }
  x0 += ks2; x1 += ks0 + 2u;
  #pragma unroll
  for (int i = 0; i < 4; ++i) { x0 += x1; x1 = rotl32(x1, ra[i]); x1 ^= x0; }
  x0 += ks0; x1 += ks1 + 3u;
  #pragma unroll
  for (int i = 0; i < 4; ++i) { x0 += x1; x1 = rotl32(x1, rb[i]); x1 ^= x0; }
  x0 += ks1; x1 += ks2 + 4u;
  #pragma unroll
  for (int i = 0; i < 4; ++i) { x0 += x1; x1 = rotl32(x1, ra[i]); x1 ^= x0; }
  x0 += ks2; x1 += ks0 + 5u;
  o0 = x0; o1 = x1;
}

// Gumbel noise matching: u = uniform(key(42),(BT,12,11,2),f32,1e-10,1); g=-log(-log(u))
static __device__ inline float gumbel_at(unsigned i) {
  const unsigned halfN = 422400u;  // (3200*12*11*2)/2
  unsigned x0, x1, o0, o1;
  if (i < halfN) { x0 = i; x1 = i + halfN; } else { x0 = i - halfN; x1 = i; }
  threefry2x32(0u, 42u, x0, x1, o0, o1);
  unsigned bits = (i < halfN) ? o0 : o1;
  unsigned fb = (bits >> 9) | 0x3f800000u;
  float u = __uint_as_float(fb) - 1.0f;            // [0,1)
  u = u * (1.0f - 1e-10f) + 1e-10f;
  u = fmaxf(u, 1e-10f);
  return -logf(-logf(u));
}

// ---------------- generic fp32 WMMA GEMM: C(Mtiles*16 x N) = A(.. x K) @ W(K x N) + bias
// A in LDS (row-major, stride lda), W global row-major, C to LDS (stride ldc).
static __device__ void gemm16(const float *A, int lda,
                              const float *__restrict__ W,
                              const float *__restrict__ bias,
                              float *C, int ldc,
                              int Mtiles, int K, int N) {
  const int lane = threadIdx.x & 31;
  const int wave = threadIdx.x >> 5;
  const int hl   = lane >> 4;      // half of wave: K-offset 0 or 2 / row-offset 0 or 8
  const int lrow = lane & 15;
  const int Ntiles = (N + 15) >> 4;
  const int total = Mtiles * Ntiles;
  for (int t = wave; t < total; t += NW) {
    const int mt = t / Ntiles, nt = t % Ntiles;
    const int col = nt * 16 + lrow;
    const bool cok = (col < N);
    const float *arow = A + (mt * 16 + lrow) * lda;
    v8f acc = {0.f, 0.f, 0.f, 0.f, 0.f, 0.f, 0.f, 0.f};
    for (int k = 0; k < K; k += 4) {
      const int ka = k + hl * 2;
      v2f av, bv;
      av.x = arow[ka];
      av.y = arow[ka + 1];
      if (cok) { bv.x = W[ka * N + col]; bv.y = W[(ka + 1) * N + col]; }
      else     { bv.x = 0.f;             bv.y = 0.f; }
      acc = __builtin_amdgcn_wmma_f32_16x16x4_f32(
          /*neg_a=*/false, av, /*neg_b=*/false, bv,
          /*c_mod=*/(short)0, acc, /*reuse_a=*/false, /*reuse_b=*/false);
    }
    if (cok) {
      const float bb = bias ? bias[col] : 0.0f;
      #pragma unroll
      for (int v = 0; v < 8; ++v)
        C[(mt * 16 + v + hl * 8) * ldc + col] = acc[v] + bb;
    }
  }
}

// LayerNorm (optionally + exact GELU) over rows in LDS; one wave per row.
static __device__ void ln_rows(float *buf, int ld, int nrows, int ncol,
                               const float *__restrict__ g,
                               const float *__restrict__ b, bool do_gelu) {
  const int lane = threadIdx.x & 31;
  const int wave = threadIdx.x >> 5;
  for (int r = wave; r < nrows; r += NW) {
    float *row = buf + r * ld;
    float s = 0.f, ss = 0.f;
    for (int c = lane; c < ncol; c += 32) { float v = row[c]; s += v; ss += v * v; }
    s = wave_sum(s); ss = wave_sum(ss);
    const float mean = s / (float)ncol;
    const float inv = rsqrtf(ss / (float)ncol - mean * mean + 1e-5f);
    for (int c = lane; c < ncol; c += 32) {
      float v = (row[c] - mean) * inv * g[c] + b[c];
      if (do_gelu) v = gelu_exact(v);
      row[c] = v;
    }
  }
}

__global__ void __launch_bounds__(NTHREADS)
qnet_kernel(Params P) {
  extern __shared__ float smem[];
  const int bt   = blockIdx.x;         // 0..3199
  const int tid  = threadIdx.x;
  const int lane = tid & 31;
  const int wave = tid >> 5;

  // ---- Phase 0: input LayerNorm of states (D=128) + actions into sact ----
  for (int r = wave; r < 16; r += NW) {
    float *dst = smem + OFF_SACT + r * LD_SACT;
    if (r < 12) {
      const float *src = P.states + ((size_t)bt * 12 + r) * 128;
      float v[4], s = 0.f, ss = 0.f;
      #pragma unroll
      for (int i = 0; i < 4; ++i) { v[i] = src[lane + 32 * i]; s += v[i]; ss += v[i] * v[i]; }
      s = wave_sum(s); ss = wave_sum(ss);
      const float mean = s / 128.f;
      const float inv = rsqrtf(ss / 128.f - mean * mean + 1e-5f);
      #pragma unroll
      for (int i = 0; i < 4; ++i) {
        const int c = lane + 32 * i;
        dst[c] = (v[i] - mean) * inv * P.fn_g[c] + P.fn_b[c];
      }
      if (lane < 16) dst[128 + lane] = P.actions[((size_t)bt * 12 + r) * 16 + lane];
    } else {
      for (int c = lane; c < 144; c += 32) dst[c] = 0.f;
    }
  }
  __syncthreads();

  // ---- se = gelu(LN(states_ln @ se_W + se_b)) ----
  gemm16(smem + OFF_SACT, LD_SACT, P.se_W, P.se_b, smem + OFF_SE, 65, 1, 128, 64);
  __syncthreads();
  ln_rows(smem + OFF_SE, 65, 12, 64, P.se_g, P.se_bb, true);
  __syncthreads();

  // ---- sae = gelu(LN([states_ln|actions] @ sae_W + sae_b)) ----
  gemm16(smem + OFF_SACT, LD_SACT, P.sae_W, P.sae_b, smem + OFF_SAE, 65, 1, 144, 64);
  __syncthreads();
  ln_rows(smem + OFF_SAE, 65, 12, 64, P.sae_g, P.sae_bb, true);
  __syncthreads();

  // ---- K, Q (from se), V (from sae) ----
  gemm16(smem + OFF_SE, 65, P.k_W, P.k_b, smem + OFF_KK, 65, 1, 64, 64);
  gemm16(smem + OFF_SE, 65, P.q_W, P.q_b, smem + OFF_QQ, 65, 1, 64, 64);
  gemm16(smem + OFF_SAE, 65, P.v_W, P.v_b, smem + OFF_VV, 65, 1, 64, 64);
  __syncthreads();

  // ---- score[h,a,jj] = q[a,h]·k[j,h] / sqrt(16), also write score output ----
  for (int e = tid; e < 528; e += NTHREADS) {
    const int h = e / 132, rem = e % 132, a = rem / 11, jj = rem % 11;
    const int j = (jj < a) ? jj : jj + 1;
    const float *qp = smem + OFF_QQ + a * 65 + h * 16;
    const float *kp = smem + OFF_KK + j * 65 + h * 16;
    float s = 0.f;
    #pragma unroll
    for (int d = 0; d < 16; ++d) s += qp[d] * kp[d];
    s *= 0.25f;  // / sqrt(dh=16)
    smem[OFF_SC + e] = s;
    P.outScore[(size_t)bt * 528 + e] = s;
  }
  __syncthreads();

  // ---- hard-attention head: qk (132x128) -> gelu(@ha1) -> @ha2 -> logits ----
  for (int e = tid; e < 144 * 128; e += NTHREADS) {
    const int r = e >> 7, c = e & 127;
    float v = 0.f;
    if (r < 132) {
      const int a = r / 11, jj = r % 11, j = (jj < a) ? jj : jj + 1;
      const int h = c >> 5, d = c & 31;
      v = (d < 16) ? smem[OFF_QQ + a * 65 + h * 16 + d]
                   : smem[OFF_KK + j * 65 + h * 16 + (d - 16)];
    }
    smem[OFF_BIG + r * 129 + c] = v;
  }
  __syncthreads();
  gemm16(smem + OFF_BIG, 129, P.ha1_W, P.ha1_b, smem + OFF_BIG + 18576, 65, 9, 128, 64);
  __syncthreads();
  for (int e = tid; e < 132 * 64; e += NTHREADS) {
    float *p = smem + OFF_BIG + 18576 + (e >> 6) * 65 + (e & 63);
    *p = gelu_exact(*p);
  }
  __syncthreads();
  gemm16(smem + OFF_BIG + 18576, 65, P.ha2_W, P.ha2_b, smem + OFF_LG, 2, 9, 64, 2);
  __syncthreads();

  // ---- Gumbel hard gate (forward value of straight-through = one-hot argmax) ----
  for (int p = tid; p < 132; p += NTHREADS) {
    const unsigned base = (unsigned)(((size_t)bt * 132 + p) * 2);
    const float z0 = smem[OFF_LG + p * 2 + 0] + gumbel_at(base);
    const float z1 = smem[OFF_LG + p * 2 + 1] + gumbel_at(base + 1);
    const float m = fmaxf(z0, z1);
    const float e0 = expf(z0 - m), e1 = expf(z1 - m);
    smem[OFF_HW + p] = (e1 / (e0 + e1) > e0 / (e0 + e1)) ? 1.f : 0.f;
  }
  __syncthreads();

  // ---- attention weights = softmax(score) * hard_w ; emit wf ----
  for (int e = tid; e < 48; e += NTHREADS) {
    const int h = e / 12, a = e % 12;
    const float *sc = smem + OFF_SC + h * 132 + a * 11;
    float mx = sc[0];
    #pragma unroll
    for (int jj = 1; jj < 11; ++jj) mx = fmaxf(mx, sc[jj]);
    float sum = 0.f;
    #pragma unroll
    for (int jj = 0; jj < 11; ++jj) sum += expf(sc[jj] - mx);
    float wrow[12];
    wrow[a] = 1.0f;
    #pragma unroll
    for (int jj = 0; jj < 11; ++jj) {
      const float w = expf(sc[jj] - mx) / sum * smem[OFF_HW + a * 11 + jj];
      smem[OFF_WT + h * 132 + a * 11 + jj] = w;
      const int j = (jj < a) ? jj : jj + 1;
      wrow[j] = w;
    }
    float *wfout = P.outWf + (((size_t)bt * 4 + h) * 12 + a) * 12;
    #pragma unroll
    for (int j2 = 0; j2 < 12; ++j2) wfout[j2] = wrow[j2];
  }
  __syncthreads();

  // ---- x = LN(sae + weighted-agg of V) ----
  for (int e = tid; e < 16 * 64; e += NTHREADS) {
    const int a = e >> 6, c = e & 63;
    float val = 0.f;
    if (a < 12) {
      const int h = c >> 4;
      float acc = 0.f;
      #pragma unroll
      for (int jj = 0; jj < 11; ++jj) {
        const int j = (jj < a) ? jj : jj + 1;
        acc += smem[OFF_WT + h * 132 + a * 11 + jj] * smem[OFF_VV + j * 65 + c];
      }
      val = smem[OFF_SAE + a * 65 + c] + acc;
    }
    smem[OFF_X + a * 65 + c] = val;
  }
  __syncthreads();
  ln_rows(smem + OFF_X, 65, 12, 64, P.avln_g, P.avln_b, false);
  __syncthreads();

  // ---- FF block: h1 = x@ff1 ; LN+gelu ; ff = h1@ff2 ; x2 = LN(x + ff) ----
  gemm16(smem + OFF_X, 65, P.ff1_W, P.ff1_b, smem + OFF_BIG, 2049, 1, 64, 2048);
  __syncthreads();
  ln_rows(smem + OFF_BIG, 2049, 12, 2048, P.ff_g, P.ff_bb, true);
  __syncthreads();
  gemm16(smem + OFF_BIG, 2049, P.ff2_W, P.ff2_b, smem + OFF_FF, 65, 1, 2048, 64);
  __syncthreads();
  for (int e = tid; e < 16 * 64; e += NTHREADS) {
    const int a = e >> 6, c = e & 63;
    smem[OFF_X2 + a * 65 + c] = smem[OFF_X + a * 65 + c] + smem[OFF_FF + a * 65 + c];
  }
  __syncthreads();
  ln_rows(smem + OFF_X2, 65, 12, 64, P.avlln_g, P.avlln_b, false);
  __syncthreads();

  // ---- critic head: c = gelu(LN([se|x2]@c1)) ; c = gelu(LN(c@c2)) ----
  for (int e = tid; e < 16 * 128; e += NTHREADS) {
    const int r = e >> 7, c = e & 127;
    smem[OFF_BIG + r * 129 + c] = (c < 64) ? smem[OFF_SE + r * 65 + c]
                                           : smem[OFF_X2 + r * 65 + (c - 64)];
  }
  __syncthreads();
  gemm16(smem + OFF_BIG, 129, P.c1_W, P.c1_b, smem + OFF_BIG + 2064, 129, 1, 128, 128);
  __syncthreads();
  ln_rows(smem + OFF_BIG + 2064, 129, 12, 128, P.c1_g, P.c1_bb, true);
  __syncthreads();
  gemm16(smem + OFF_BIG + 2064, 129, P.c2_W, P.c2_b, smem + OFF_C2, 65, 1, 128, 64);
  __syncthreads();
  ln_rows(smem + OFF_C2, 65, 12, 64, P.c2_g, P.c2_bb, true);
  __syncthreads();

  // ---- qv = LN(c, qln) @ qv_W + qv_b ; Q = sum(actions * qv) ----
  for (int e = tid; e < 16 * 64; e += NTHREADS)
    smem[OFF_QC + (e >> 6) * 65 + (e & 63)] = smem[OFF_C2 + (e >> 6) * 65 + (e & 63)];
  __syncthreads();
  ln_rows(smem + OFF_QC, 65, 12, 64, P.qln_g, P.qln_b, false);
  __syncthreads();
  gemm16(smem + OFF_QC, 65, P.qv_W, P.qv_b, smem + OFF_QV, 17, 1, 64, 16);
  __syncthreads();
  for (int a = tid; a < 12; a += NTHREADS) {
    float q = 0.f;
    #pragma unroll
    for (int n = 0; n < 16; ++n)
      q += smem[OFF_SACT + a * LD_SACT + 128 + n] * smem[OFF_QV + a * 17 + n];
    P.outQ[(size_t)bt * 12 + a] = q;
  }
}

extern "C" void kernel_launch(void* const* d_in, const int* in_sizes, int n_in,
                              void* d_out, int out_size, void* d_ws, size_t ws_size,
                              hipStream_t stream) {
  (void)in_sizes; (void)n_in; (void)d_ws; (void)ws_size; (void)out_size;
  Params P;
  int i = 0;
  P.states  = (const float*)d_in[i++]; P.actions = (const float*)d_in[i++];
  P.fn_g    = (const float*)d_in[i++]; P.fn_b    = (const float*)d_in[i++];
  P.se_W    = (const float*)d_in[i++]; P.se_b    = (const float*)d_in[i++];
  P.se_g    = (const float*)d_in[i++]; P.se_bb   = (const float*)d_in[i++];
  P.sae_W   = (const float*)d_in[i++]; P.sae_b   = (const float*)d_in[i++];
  P.sae_g   = (const float*)d_in[i++]; P.sae_bb  = (const float*)d_in[i++];
  P.k_W     = (const float*)d_in[i++]; P.k_b     = (const float*)d_in[i++];
  P.q_W     = (const float*)d_in[i++]; P.q_b     = (const float*)d_in[i++];
  P.v_W     = (const float*)d_in[i++]; P.v_b     = (const float*)d_in[i++];
  P.avln_g  = (const float*)d_in[i++]; P.avln_b  = (const float*)d_in[i++];
  P.ff1_W   = (const float*)d_in[i++]; P.ff1_b   = (const float*)d_in[i++];
  P.ff_g    = (const float*)d_in[i++]; P.ff_bb   = (const float*)d_in[i++];
  P.ff2_W   = (const float*)d_in[i++]; P.ff2_b   = (const float*)d_in[i++];
  P.avlln_g = (const float*)d_in[i++]; P.avlln_b = (const float*)d_in[i++];
  P.ha1_W   = (const float*)d_in[i++]; P.ha1_b   = (const float*)d_in[i++];
  P.ha2_W   = (const float*)d_in[i++]; P.ha2_b   = (const float*)d_in[i++];
  P.c1_W    = (const float*)d_in[i++]; P.c1_b    = (const float*)d_in[i++];
  P.c1_g    = (const float*)d_in[i++]; P.c1_bb   = (const float*)d_in[i++];
  P.c2_W    = (const float*)d_in[i++]; P.c2_b    = (const float*)d_in[i++];
  P.c2_g    = (const float*)d_in[i++]; P.c2_bb   = (const float*)d_in[i++];
  P.qln_g   = (const float*)d_in[i++]; P.qln_b   = (const float*)d_in[i++];
  P.qv_W    = (const float*)d_in[i++]; P.qv_b    = (const float*)d_in[i++];

  float* out = (float*)d_out;
  P.outQ     = out;                              // 3200*12
  P.outWf    = out + 3200 * 12;                  // 3200*4*12*12
  P.outScore = out + 3200 * 12 + 3200 * 4 * 12 * 12;  // 3200*4*12*11

  const size_t shmem = (size_t)SMEM_FLOATS * sizeof(float);
  hipFuncSetAttribute((const void*)qnet_kernel,
                      hipFuncAttributeMaxDynamicSharedMemorySize, (int)shmem);
  qnet_kernel<<<3200, NTHREADS, shmem, stream>>>(P);
}